// SLAM_Layer_48421461295188
// MI455X (gfx1250) — compile-verified
//
#include <hip/hip_runtime.h>

typedef __attribute__((ext_vector_type(2))) float v2f;
typedef __attribute__((ext_vector_type(8))) float v8f;

#define T_DIM  128
#define M_DIM  512
#define K_DIM  256   // contraction dim of stage-2 / rows-of-aux "k"
#define DA     256   // aux feature dim (stage-1 contraction)
#define DENSE  512   // output feature dim "e"

// D = A(16x4) * B(4x16) + C, f32 exact. reuse hints must be false (instrs differ).
#define WMMA_F32(a, b, c) \
    __builtin_amdgcn_wmma_f32_16x16x4_f32(false, (a), false, (b), (short)0, (c), false, false)

// A fragment: 16x4 (MxK) f32, row-major source with row stride lda.
// lane 0-15: M=lane, K={k0,k0+1}; lane 16-31: M=lane-16, K={k0+2,k0+3}
__device__ __forceinline__ v2f load_a_frag(const float* __restrict__ A, int lda,
                                           int mrow, int k0, int lane) {
    const int m  = mrow + (lane & 15);
    const int kk = k0 + ((lane >> 4) << 1);
    return *(const v2f*)(A + m * lda + kk);   // 8B-aligned (k0 multiple of 4)
}

// B fragment: 4x16 (KxN) f32, row-major source with row stride ldb.
// lane 0-15: N=ncol+lane, K={k0,k0+1}; lane 16-31: N=ncol+lane-16, K={k0+2,k0+3}
__device__ __forceinline__ v2f load_b_frag(const float* __restrict__ B, int ldb,
                                           int k0, int ncol, int lane) {
    const int n  = ncol + (lane & 15);
    const int kk = k0 + ((lane >> 4) << 1);
    v2f b;
    b.x = B[kk * ldb + n];
    b.y = B[(kk + 1) * ldb + n];
    return b;
}

// ---------------------------------------------------------------------------
// Stage 1: aux_out[t,k,e] = relu(x_aux[k,t,:] . W[:,e] + b[e])
// Treated as GEMM: rows r = k*T + t (x_aux is [K,T,DA] row-major, DA contig),
// cols e in [0,512). One wave computes a 32-row x 64-col tile.
// Output stored TRANSPOSED into ws as [T, K, DENSE].
// ---------------------------------------------------------------------------
__global__ __launch_bounds__(32)
void slam_aux_dense_wmma(const float* __restrict__ x_aux,
                         const float* __restrict__ W,
                         const float* __restrict__ bias,
                         float* __restrict__ ws) {
    const int lane = threadIdx.x & 31;
    const int tile = blockIdx.x;        // tm * 8 + tn
    const int tn   = tile & 7;          // 64-col tile -> col0
    const int tm   = tile >> 3;         // 32-row tile -> row0
    const int row0 = tm * 32;           // global row in [K_DIM * T_DIM)
    const int col0 = tn * 64;

    const v8f zero = {};
    v8f acc[2][4];
    #pragma unroll
    for (int i = 0; i < 2; ++i)
        #pragma unroll
        for (int j = 0; j < 4; ++j)
            acc[i][j] = zero;

    const float* A = x_aux + (size_t)row0 * DA;

    for (int k0 = 0; k0 < DA; k0 += 4) {
        v2f a0 = load_a_frag(A, DA, 0,  k0, lane);
        v2f a1 = load_a_frag(A, DA, 16, k0, lane);
        v2f b0 = load_b_frag(W, DENSE, k0, col0,      lane);
        v2f b1 = load_b_frag(W, DENSE, k0, col0 + 16, lane);
        v2f b2 = load_b_frag(W, DENSE, k0, col0 + 32, lane);
        v2f b3 = load_b_frag(W, DENSE, k0, col0 + 48, lane);

        acc[0][0] = WMMA_F32(a0, b0, acc[0][0]);
        acc[0][1] = WMMA_F32(a0, b1, acc[0][1]);
        acc[0][2] = WMMA_F32(a0, b2, acc[0][2]);
        acc[0][3] = WMMA_F32(a0, b3, acc[0][3]);
        acc[1][0] = WMMA_F32(a1, b0, acc[1][0]);
        acc[1][1] = WMMA_F32(a1, b1, acc[1][1]);
        acc[1][2] = WMMA_F32(a1, b2, acc[1][2]);
        acc[1][3] = WMMA_F32(a1, b3, acc[1][3]);
    }

    // Epilogue: bias + relu, scatter-store transposed to ws[t][k][e].
    // Within this 32-row tile k is constant (128 % 32 == 0): r = k*128 + t.
    const int k_idx = row0 >> 7;        // r / 128
    const int t0    = row0 & 127;       // r % 128
    const int mOff  = (lane >> 4) * 8;  // C layout: lanes 16-31 hold M=8..15
    const int nlane = lane & 15;

    #pragma unroll
    for (int mi = 0; mi < 2; ++mi) {
        #pragma unroll
        for (int ni = 0; ni < 4; ++ni) {
            const int   n  = col0 + ni * 16 + nlane;
            const float bv = bias[n];
            #pragma unroll
            for (int v = 0; v < 8; ++v) {
                const int mloc = mi * 16 + mOff + v;     // 0..31 within tile
                const int t    = t0 + mloc;
                float val = acc[mi][ni][v] + bv;
                val = fmaxf(val, 0.0f);
                ws[((size_t)t * K_DIM + k_idx) * DENSE + n] = val;
            }
        }
    }
}

// ---------------------------------------------------------------------------
// Stage 2: out[t,m,e] = sum_k main[t,m,k] * ws[t,k,e]
// 128 independent 512x256x512 GEMMs. One wave per 32x64 tile of out[t].
// ---------------------------------------------------------------------------
__global__ __launch_bounds__(32)
void slam_batched_gemm_wmma(const float* __restrict__ x_main,  // [T,M,K]
                            const float* __restrict__ ws,       // [T,K,DENSE]
                            float* __restrict__ out) {          // [T,M,DENSE]
    const int lane = threadIdx.x & 31;
    const int idx  = blockIdx.x;        // t * 128 + tm * 8 + tn
    const int t    = idx >> 7;
    const int rem  = idx & 127;
    const int tm   = rem >> 3;          // 0..15 : 32-row tile
    const int tn   = rem & 7;           // 0..7  : 64-col tile
    const int col0 = tn * 64;

    const float* A = x_main + (size_t)t * M_DIM * K_DIM + (size_t)tm * 32 * K_DIM;
    const float* B = ws     + (size_t)t * K_DIM * DENSE;
    float*       C = out    + (size_t)t * M_DIM * DENSE;

    const v8f zero = {};
    v8f acc[2][4];
    #pragma unroll
    for (int i = 0; i < 2; ++i)
        #pragma unroll
        for (int j = 0; j < 4; ++j)
            acc[i][j] = zero;

    for (int k0 = 0; k0 < K_DIM; k0 += 4) {
        v2f a0 = load_a_frag(A, K_DIM, 0,  k0, lane);
        v2f a1 = load_a_frag(A, K_DIM, 16, k0, lane);
        v2f b0 = load_b_frag(B, DENSE, k0, col0,      lane);
        v2f b1 = load_b_frag(B, DENSE, k0, col0 + 16, lane);
        v2f b2 = load_b_frag(B, DENSE, k0, col0 + 32, lane);
        v2f b3 = load_b_frag(B, DENSE, k0, col0 + 48, lane);

        acc[0][0] = WMMA_F32(a0, b0, acc[0][0]);
        acc[0][1] = WMMA_F32(a0, b1, acc[0][1]);
        acc[0][2] = WMMA_F32(a0, b2, acc[0][2]);
        acc[0][3] = WMMA_F32(a0, b3, acc[0][3]);
        acc[1][0] = WMMA_F32(a1, b0, acc[1][0]);
        acc[1][1] = WMMA_F32(a1, b1, acc[1][1]);
        acc[1][2] = WMMA_F32(a1, b2, acc[1][2]);
        acc[1][3] = WMMA_F32(a1, b3, acc[1][3]);
    }

    const int mOff  = (lane >> 4) * 8;
    const int nlane = lane & 15;
    #pragma unroll
    for (int mi = 0; mi < 2; ++mi) {
        #pragma unroll
        for (int ni = 0; ni < 4; ++ni) {
            const int n = col0 + ni * 16 + nlane;
            #pragma unroll
            for (int v = 0; v < 8; ++v) {
                const int m = tm * 32 + mi * 16 + mOff + v;
                C[(size_t)m * DENSE + n] = acc[mi][ni][v];
            }
        }
    }
}

extern "C" void kernel_launch(void* const* d_in, const int* in_sizes, int n_in,
                              void* d_out, int out_size, void* d_ws, size_t ws_size,
                              hipStream_t stream) {
    const float* x_main = (const float*)d_in[0];  // [1,128,512,256]
    const float* x_aux  = (const float*)d_in[1];  // [256,128,256]
    const float* W      = (const float*)d_in[2];  // [256,512]
    const float* b      = (const float*)d_in[3];  // [512]
    float*       out    = (float*)d_out;          // [1,128,512,512,1]
    float*       ws     = (float*)d_ws;           // aux_out transposed [128,256,512] = 64 MB

    // Stage 1: (K*T)/32 row-tiles x DENSE/64 col-tiles = 1024 * 8 waves
    const int grid1 = (K_DIM * T_DIM / 32) * (DENSE / 64);
    slam_aux_dense_wmma<<<grid1, 32, 0, stream>>>(x_aux, W, b, ws);

    // Stage 2: T * (M/32) * (DENSE/64) = 128 * 16 * 8 waves
    const int grid2 = T_DIM * (M_DIM / 32) * (DENSE / 64);
    slam_batched_gemm_wmma<<<grid2, 32, 0, stream>>>(x_main, ws, out);
}